// GCN_24850680775355
// MI455X (gfx1250) — compile-verified
//
#include <hip/hip_runtime.h>
#include <math.h>

// ---- problem constants -------------------------------------------------
#define BB 4
#define FF 16
#define OO 15
#define NB_OBJ 100
#define NB_VERB 157
#define D_EMB 2048
#define N_TOP 10
#define BF (BB*FF)            // 64
#define FM_LD 2248            // fm row stride (we use first 2048 cols)

typedef __attribute__((ext_vector_type(2))) float v2f;
typedef __attribute__((ext_vector_type(8))) float v8f;

// ------------------------------------------------------------------------
// Y[M x 2048] = X[M x 2048 (row stride LDX)] @ W^T (+ bias).  fp32 WMMA.
// Wave tile: 64(M) x 64(N) = 16 accumulator tiles (128 VGPRs).
// Per k-step of 4: 4 A-frag + 4 B-frag b64 loads feed 16 WMMAs (0.5 ld/wmma).
// LDX is a template constant so the 16-row stride folds into the 24-bit
// instruction offset (2 live address registers total).
// A 16x4 f32 layout: lane l -> row (l&15), k-half (l>>4)*2, float2 {k,k+1}.
// B 4x16 f32 layout: lane l -> col (l&15), same k-half, float2 from W row.
// D 16x16 f32 layout: acc[r] holds (row r + 8*(l>>4), col l&15).
// Block = 4 waves side by side in N: block tile 64(M) x 256(N).
// ------------------------------------------------------------------------
template<int LDX>
__global__ __launch_bounds__(128)
void gemm_xwT(const float* __restrict__ X, const float* __restrict__ W,
              const float* __restrict__ bias, float* __restrict__ Y)
{
    const int lane = threadIdx.x & 31;
    const int wave = threadIdx.x >> 5;
    const int ml   = lane & 15;            // A row / B col / D col
    const int kh   = (lane >> 4) << 1;     // 0 or 2
    const int m0   = blockIdx.y * 64;
    const int e0   = blockIdx.x * 256 + wave * 64;

    const float* xp = X + (size_t)(m0 + ml) * LDX   + kh;
    const float* wp = W + (size_t)(e0 + ml) * D_EMB + kh;

    v8f acc[4][4] = {};

    #pragma unroll 2
    for (int k = 0; k < D_EMB; k += 4) {
        v2f a[4], b[4];
        #pragma unroll
        for (int i = 0; i < 4; ++i)
            a[i] = *(const v2f*)(xp + (size_t)(i * 16) * LDX + k);
        #pragma unroll
        for (int j = 0; j < 4; ++j)
            b[j] = *(const v2f*)(wp + (size_t)(j * 16) * D_EMB + k);
        #pragma unroll
        for (int i = 0; i < 4; ++i)
            #pragma unroll
            for (int j = 0; j < 4; ++j)
                acc[i][j] = __builtin_amdgcn_wmma_f32_16x16x4_f32(
                    false, a[i], false, b[j], (short)0, acc[i][j], false, false);
    }

    float bv[4];
    #pragma unroll
    for (int j = 0; j < 4; ++j)
        bv[j] = bias ? bias[e0 + j * 16 + ml] : 0.0f;

    const int rbase = m0 + ((lane >> 4) ? 8 : 0);
    #pragma unroll
    for (int i = 0; i < 4; ++i) {
        #pragma unroll
        for (int r = 0; r < 8; ++r) {
            float* yr = Y + (size_t)(rbase + i * 16 + r) * D_EMB + e0 + ml;
            #pragma unroll
            for (int j = 0; j < 4; ++j)
                yr[j * 16] = acc[i][j][r] + bv[j];
        }
    }
}

// P = A_v2o(100x157) @ A_o2v(157x100); s[c] = sum_v A_v2o[c,v]
__global__ void make_P(const float* __restrict__ A_v2o,
                       const float* __restrict__ A_o2v,
                       float* __restrict__ P, float* __restrict__ s)
{
    int idx = blockIdx.x * 256 + threadIdx.x;
    if (idx < NB_OBJ * NB_OBJ) {
        int c2 = idx % NB_OBJ, c = idx / NB_OBJ;
        float acc = 0.f;
        #pragma unroll 1
        for (int v = 0; v < NB_VERB; ++v)
            acc += A_v2o[c * NB_VERB + v] * A_o2v[v * NB_OBJ + c2];
        P[idx] = acc;
    }
    if (idx < NB_OBJ) {
        float acc = 0.f;
        #pragma unroll 1
        for (int v = 0; v < NB_VERB; ++v) acc += A_v2o[idx * NB_VERB + v];
        s[idx] = acc;
    }
}

// grid (D_EMB/256, NB_OBJ, BF):  x[bf,c,d] = sum_o scores[bf,o,c]*g[bf*15+o,d] + b_obj[d]
__global__ void frame_mix(const float* __restrict__ g,
                          const float* __restrict__ scores,
                          const float* __restrict__ b_obj,
                          float* __restrict__ x)
{
    const int d  = blockIdx.x * 256 + threadIdx.x;
    const int c  = blockIdx.y;
    const int bf = blockIdx.z;
    const float* sp = scores + bf * (OO * NB_OBJ) + c;
    const float* gp = g + (size_t)bf * (OO * D_EMB) + d;
    float acc = b_obj[d];
    #pragma unroll
    for (int o = 0; o < OO; ++o)
        acc += sp[o * NB_OBJ] * gp[o * D_EMB];
    x[((size_t)bf * NB_OBJ + c) * D_EMB + d] = acc;
}

// grid (D_EMB/256, NB_OBJ, BF):  z[bf,c,d] = sum_{c2} P[c,c2]*y[bf,c2,d] + s[c]*b1[d]
__global__ void z_mix(const float* __restrict__ P, const float* __restrict__ s,
                      const float* __restrict__ y, const float* __restrict__ b1,
                      float* __restrict__ z)
{
    const int d  = blockIdx.x * 256 + threadIdx.x;
    const int c  = blockIdx.y;
    const int bf = blockIdx.z;
    const float* yp = y + (size_t)bf * (NB_OBJ * D_EMB) + d;
    const float* Pr = P + c * NB_OBJ;
    float acc = s[c] * b1[d];
    #pragma unroll 4
    for (int c2 = 0; c2 < NB_OBJ; ++c2)
        acc += Pr[c2] * yp[(size_t)c2 * D_EMB];
    z[((size_t)bf * NB_OBJ + c) * D_EMB + d] = acc;
}

// Per (b,f): row-sums over d, top-10 (ties -> lower index, like lax.top_k),
// gather rows into out, write indices (as float) after the gathered block.
__global__ void topk_gather(const float* __restrict__ x,
                            float* __restrict__ out_ma,
                            float* __restrict__ out_idx)
{
    __shared__ float act[NB_OBJ];
    __shared__ int   sidx[N_TOP];
    const int bf = blockIdx.x;
    const int t  = threadIdx.x;            // 128 threads
    const float* xb = x + (size_t)bf * NB_OBJ * D_EMB;

    if (t < NB_OBJ) {
        const float* row = xb + (size_t)t * D_EMB;
        float a = 0.f;
        #pragma unroll 4
        for (int d = 0; d < D_EMB; ++d) a += row[d];
        act[t] = a;
    }
    __syncthreads();
    if (t == 0) {
        unsigned long long used0 = 0ull, used1 = 0ull;   // 100-bit mask
        #pragma unroll 1
        for (int j = 0; j < N_TOP; ++j) {
            int best = 0; float bv = -INFINITY;
            #pragma unroll 1
            for (int c = 0; c < NB_OBJ; ++c) {
                bool u = (c < 64) ? ((used0 >> c) & 1ull)
                                  : ((used1 >> (c - 64)) & 1ull);
                if (!u && act[c] > bv) { bv = act[c]; best = c; }
            }
            if (best < 64) used0 |= 1ull << best;
            else           used1 |= 1ull << (best - 64);
            sidx[j] = best;
        }
    }
    __syncthreads();
    #pragma unroll 1
    for (int j = 0; j < N_TOP; ++j) {
        const float* row = xb + (size_t)sidx[j] * D_EMB;
        float* dst = out_ma + ((size_t)bf * N_TOP + j) * D_EMB;
        for (int d = t; d < D_EMB; d += 128) dst[d] = row[d];
    }
    if (t < N_TOP) out_idx[bf * N_TOP + t] = (float)sidx[t];
}

// ------------------------------------------------------------------------
extern "C" void kernel_launch(void* const* d_in, const int* in_sizes, int n_in,
                              void* d_out, int out_size, void* d_ws, size_t ws_size,
                              hipStream_t stream)
{
    (void)in_sizes; (void)n_in; (void)out_size; (void)ws_size;
    const float* fm     = (const float*)d_in[0];
    const float* scores = (const float*)d_in[1];
    const float* A_o2v  = (const float*)d_in[2];
    const float* A_v2o  = (const float*)d_in[3];
    const float* W_obj  = (const float*)d_in[4];
    const float* b_obj  = (const float*)d_in[5];
    const float* W_o2v  = (const float*)d_in[6];   // [2][2048][2048]
    const float* b_o2v  = (const float*)d_in[7];   // [2][2048]
    const float* W_v2o  = (const float*)d_in[8];
    const float* b_v2o  = (const float*)d_in[9];

    char* ws = (char*)d_ws;
    const size_t szBig = (size_t)BF * NB_OBJ * D_EMB * sizeof(float);   // 52.4 MB
    float* bufA = (float*)(ws);                    // x
    float* bufB = (float*)(ws + szBig);            // y
    float* bufC = (float*)(ws + 2 * szBig);        // z
    float* g    = bufC;   // 960x2048, dead before bufC is first written
    float* P    = (float*)(ws + 3 * szBig);
    float* s    = P + NB_OBJ * NB_OBJ;

    const dim3 blk(128, 1, 1);
    const dim3 mixGrid(D_EMB / 256, NB_OBJ, BF);   // (8, 100, 64), no div/mod

    // 1) g[960x2048] = obj_feats @ W_obj^T  (fm rows, stride 2248, first 2048 cols)
    gemm_xwT<FM_LD><<<dim3(D_EMB / 256, (BF * OO) / 64), blk, 0, stream>>>(fm, W_obj, nullptr, g);
    // 2) P = A_v2o @ A_o2v ; s = rowsum(A_v2o)
    make_P<<<(NB_OBJ * NB_OBJ + 255) / 256, 256, 0, stream>>>(A_v2o, A_o2v, P, s);
    // 3) x = scores^T @ g + b_obj
    frame_mix<<<mixGrid, 256, 0, stream>>>(g, scores, b_obj, bufA);
    // 4) two layers: y = x@W1^T ; z = P@y + s*b1 ; x = z@W2^T + b2
    for (int l = 0; l < 2; ++l) {
        const float* W1 = W_o2v + (size_t)l * D_EMB * D_EMB;
        const float* b1 = b_o2v + (size_t)l * D_EMB;
        const float* W2 = W_v2o + (size_t)l * D_EMB * D_EMB;
        const float* b2 = b_v2o + (size_t)l * D_EMB;
        gemm_xwT<D_EMB><<<dim3(D_EMB / 256, (BF * NB_OBJ) / 64), blk, 0, stream>>>(bufA, W1, nullptr, bufB);
        z_mix<<<mixGrid, 256, 0, stream>>>(P, s, bufB, b1, bufC);
        gemm_xwT<D_EMB><<<dim3(D_EMB / 256, (BF * NB_OBJ) / 64), blk, 0, stream>>>(bufC, W2, b2, bufA);
    }
    // 5) row-sum, top-10, gather; indices written as float after the gathered block
    topk_gather<<<BF, 128, 0, stream>>>(bufA, (float*)d_out,
                                        (float*)d_out + (size_t)BF * N_TOP * D_EMB);
}